// HelmholtzEquation_56032143344265
// MI455X (gfx1250) — compile-verified
//
#include <hip/hip_runtime.h>

#define NN     96
#define N2     (NN * NN)
#define GG     2
#define DXH    (6.0f / 95.0f)
#define NITERS 600
#define TPB    512            // 16 wave32s on the WGP's 4 SIMD32s
#define NWAVES (TPB / 32)

typedef __attribute__((ext_vector_type(2))) float v2f;
typedef __attribute__((ext_vector_type(8))) float v8f;

// ---------------------------------------------------------------------------
// Setup: evaluate gaussian coefficient maps, drift terms, RHS on the grid.
// ---------------------------------------------------------------------------
__device__ __forceinline__ float gmap(float x, float y, const float* __restrict__ p) {
  float s = 1.0f;
#pragma unroll
  for (int g = 0; g < GG; ++g) {
    float dx = x - p[GG + g];       // cx
    float dy = y - p[2 * GG + g];   // cy
    s += p[g] * __expf(-(dx * dx + dy * dy));
  }
  return s;
}

__global__ void helm_setup_kernel(const float* __restrict__ alpha,
                                  const float* __restrict__ kp,
                                  float* __restrict__ dmap,
                                  float* __restrict__ kterm,
                                  float* __restrict__ ddx,
                                  float* __restrict__ ddy,
                                  float* __restrict__ rhs) {
  int idx = blockIdx.x * blockDim.x + threadIdx.x;
  if (idx >= N2) return;
  int i = idx / NN, j = idx % NN;
  float x = -3.0f + DXH * (float)i;
  float y = -3.0f + DXH * (float)j;

  float d  = gmap(x, y, alpha);
  float km = gmap(x, y, kp);
  dmap[idx]  = d;
  kterm[idx] = km * km * (DXH * DXH);

  bool interior = (i > 0) && (i < NN - 1) && (j > 0) && (j < NN - 1);
  float gx = 0.0f, gy = 0.0f;
  if (interior) {
    // d_diff_x = (A @ D)/2 = (D[i-1,j] - D[i+1,j])/2 ; masked to interior
    gx = 0.5f * (gmap(x - DXH, y, alpha) - gmap(x + DXH, y, alpha));
    gy = 0.5f * (gmap(x, y - DXH, alpha) - gmap(x, y + DXH, alpha));
  }
  ddx[idx] = gx;
  ddy[idx] = gy;

  const float PI = 3.14159265358979323846f;
  rhs[idx] = __sinf(PI * x) * __sinf(PI * y) * (DXH * DXH);
}

// ---------------------------------------------------------------------------
// WMMA f32 16x16x4 helper. A: 16x4, B: 4x16, C/D: 16x16 f32.
// ---------------------------------------------------------------------------
__device__ __forceinline__ v8f wmma4(v2f a, v2f b, v8f c) {
  return __builtin_amdgcn_wmma_f32_16x16x4_f32(false, a, false, b, (short)0, c,
                                               false, false);
}

// Operator application:  Y = -( d .* (L1 U + U L1) + kterm .* U
//                               + 0.5*ddx .* (A U) + 0.5*ddy .* (U A^T) )
// L1/A operand tiles synthesized from shared indicator values (entries in
// {-2,-1,0,1}); K loop band-limited to the diagonal block plus 4 k on either side.
__device__ void helm_matvec(const float* __restrict__ Uin, float* __restrict__ Yout,
                            const float* __restrict__ dmap,
                            const float* __restrict__ kterm,
                            const float* __restrict__ ddx,
                            const float* __restrict__ ddy,
                            int wave, int lane) {
  const int h   = lane >> 4;   // half-wave: 0 or 1
  const int l16 = lane & 15;

  for (int t = wave; t < 36; t += NWAVES) {   // 6x6 tiles of 16x16
    const int bi = t / 6, bj = t % 6;
    const int m = 16 * bi + l16;   // A-operand row for left products
    const int n = 16 * bj + l16;   // B-operand col

    v8f aD2L = {}; // L1 @ U
    v8f aAx  = {}; // A  @ U
    v8f aD2R = {}; // U @ L1
    v8f aAy  = {}; // U @ A^T

    // ---- left products: band of K around bi ----
    int klo = 16 * bi - 4; if (klo < 0) klo = 0;
    int khi = 16 * bi + 20; if (khi > NN) khi = NN;
    for (int kk = klo; kk < khi; kk += 4) {
      v2f aL1, aA, bU;
#pragma unroll
      for (int v = 0; v < 2; ++v) {
        int k = kk + 2 * h + v;
        float em1 = (k == m - 1) ? 1.0f : 0.0f;
        float e0  = (k == m)     ? 1.0f : 0.0f;
        float ep1 = (k == m + 1) ? 1.0f : 0.0f;
        aL1[v] = em1 + ep1 - 2.0f * e0;
        aA[v]  = em1 - ep1;
        bU[v]  = Uin[k * NN + n];
      }
      aD2L = wmma4(aL1, bU, aD2L);
      aAx  = wmma4(aA,  bU, aAx);
    }

    // ---- right products: band of K around bj ----
    klo = 16 * bj - 4; if (klo < 0) klo = 0;
    khi = 16 * bj + 20; if (khi > NN) khi = NN;
    for (int kk = klo; kk < khi; kk += 4) {
      v2f aU, bL1, bAt;
#pragma unroll
      for (int v = 0; v < 2; ++v) {
        int k = kk + 2 * h + v;
        float em1 = (k == n - 1) ? 1.0f : 0.0f;
        float e0  = (k == n)     ? 1.0f : 0.0f;
        float ep1 = (k == n + 1) ? 1.0f : 0.0f;
        aU[v]  = Uin[m * NN + k];
        bL1[v] = em1 + ep1 - 2.0f * e0;   // L1[k, n]
        bAt[v] = em1 - ep1;               // A^T[k, n] = A[n, k]
      }
      aD2R = wmma4(aU, bL1, aD2R);
      aAy  = wmma4(aU, bAt, aAy);
    }

    // ---- pointwise combine, write tile ----
#pragma unroll
    for (int v = 0; v < 8; ++v) {
      int i = 16 * bi + v + 8 * h;   // C/D layout: VGPR v holds rows v / v+8
      int j = 16 * bj + l16;
      int idx = i * NN + j;
      float u = Uin[idx];
      Yout[idx] = -(dmap[idx] * (aD2L[v] + aD2R[v]) + kterm[idx] * u +
                    0.5f * ddx[idx] * aAx[v] + 0.5f * ddy[idx] * aAy[v]);
    }
  }
}

__device__ __forceinline__ float guard(float d) {
  float a = fabsf(d);
  return (a > 1e-30f) ? d : ((d >= 0.0f) ? 1e-30f : -1e-30f);
}

__device__ float block_dot(const float* __restrict__ a, const float* __restrict__ b,
                           float* red, int tid) {
  float s = 0.0f;
  for (int i = tid; i < N2; i += TPB) s += a[i] * b[i];
  red[tid] = s;
  __syncthreads();
  for (int off = TPB / 2; off > 0; off >>= 1) {
    if (tid < off) red[tid] += red[tid + off];
    __syncthreads();
  }
  float r = red[0];
  __syncthreads();
  return r;
}

// ---------------------------------------------------------------------------
// Single-workgroup BiCGStab; all 7 solver vectors resident in ~254 KB LDS.
// ---------------------------------------------------------------------------
__launch_bounds__(TPB)
__global__ void helm_bicgstab_kernel(const float* __restrict__ dmap,
                                     const float* __restrict__ kterm,
                                     const float* __restrict__ ddx,
                                     const float* __restrict__ ddy,
                                     const float* __restrict__ rhs,
                                     float* __restrict__ uout) {
  extern __shared__ float lds[];
  float* x   = lds + 0 * N2;
  float* r   = lds + 1 * N2;
  float* rh  = lds + 2 * N2;
  float* p   = lds + 3 * N2;
  float* vv  = lds + 4 * N2;
  float* s   = lds + 5 * N2;
  float* tt  = lds + 6 * N2;
  float* red = lds + 7 * N2;   // TPB floats

  const int tid  = threadIdx.x;
  const int wave = tid >> 5;
  const int lane = tid & 31;

  for (int i = tid; i < N2; i += TPB) {
    float b = rhs[i];
    x[i] = 0.0f; r[i] = b; rh[i] = b; p[i] = 0.0f; vv[i] = 0.0f;
  }
  __syncthreads();

  float rho_old = 1.0f, alpha = 1.0f, omega = 1.0f;

  for (int it = 0; it < NITERS; ++it) {
    float rho  = block_dot(rh, r, red, tid);
    float beta = (rho / guard(rho_old)) * (alpha / guard(omega));
    for (int i = tid; i < N2; i += TPB) p[i] = r[i] + beta * (p[i] - omega * vv[i]);
    __syncthreads();

    helm_matvec(p, vv, dmap, kterm, ddx, ddy, wave, lane);
    __syncthreads();

    float rv = block_dot(rh, vv, red, tid);
    alpha = rho / guard(rv);
    for (int i = tid; i < N2; i += TPB) s[i] = r[i] - alpha * vv[i];
    __syncthreads();

    helm_matvec(s, tt, dmap, kterm, ddx, ddy, wave, lane);
    __syncthreads();

    float ts = block_dot(tt, s, red, tid);
    float t2 = block_dot(tt, tt, red, tid);
    omega = ts / guard(t2);
    for (int i = tid; i < N2; i += TPB) {
      x[i] += alpha * p[i] + omega * s[i];
      r[i]  = s[i] - omega * tt[i];
    }
    __syncthreads();
    rho_old = rho;
  }

  for (int i = tid; i < N2; i += TPB) uout[i] = x[i];
}

// ---------------------------------------------------------------------------
extern "C" void kernel_launch(void* const* d_in, const int* in_sizes, int n_in,
                              void* d_out, int out_size, void* d_ws, size_t ws_size,
                              hipStream_t stream) {
  (void)in_sizes; (void)n_in; (void)out_size; (void)ws_size;
  const float* alpha = (const float*)d_in[0];
  const float* kp    = (const float*)d_in[1];

  float* ws    = (float*)d_ws;
  float* dmap  = ws + 0 * N2;
  float* kterm = ws + 1 * N2;
  float* ddx   = ws + 2 * N2;
  float* ddy   = ws + 3 * N2;
  float* rhs   = ws + 4 * N2;

  helm_setup_kernel<<<(N2 + 255) / 256, 256, 0, stream>>>(
      alpha, kp, dmap, kterm, ddx, ddy, rhs);

  const size_t shbytes = (size_t)(7 * N2 + TPB) * sizeof(float);  // ~254 KB < 320 KB
  (void)hipFuncSetAttribute((const void*)helm_bicgstab_kernel,
                            hipFuncAttributeMaxDynamicSharedMemorySize,
                            (int)shbytes);
  helm_bicgstab_kernel<<<1, TPB, shbytes, stream>>>(
      dmap, kterm, ddx, ddy, rhs, (float*)d_out);
}